// RMSNorm_37855841747142
// MI455X (gfx1250) — compile-verified
//
#include <hip/hip_runtime.h>

// RMSNorm: out[r, :] = in[r, :] / sqrt(mean(in[r,:]^2) + EPS) * gamma[r, :]
// B*S = 16384 rows, D = 2048 fp32. Pure HBM-bandwidth-bound streaming kernel:
// 12 B/elem * 32Mi elem = 402 MB  ->  ~17 us at 23.3 TB/s.
// CDNA5 paths used: global_load_async_to_lds_b128 (ASYNCcnt DMA), wave32
// shuffle reduction, non-temporal load/store hints (working set >> 192MB L2).

#define RMS_EPS 5e-06f
#define ROW_D   2048
#define NTHREADS 256   // 8 wave32 per block; 8 floats per thread

typedef __attribute__((ext_vector_type(4))) float v4f;

__global__ __launch_bounds__(NTHREADS) void rmsnorm_f32_kernel(
    const float* __restrict__ in,
    const float* __restrict__ gamma,
    float* __restrict__ out)
{
    __shared__ float row[ROW_D];        // 8 KB staging for one row
    __shared__ float partials[NTHREADS / 32];

    const int tid = threadIdx.x;
    const long long rbase = (long long)blockIdx.x * ROW_D;
    const float* in_row = in    + rbase;
    const float* g_row  = gamma + rbase;
    float*       o_row  = out   + rbase;

    // LDS byte address of this thread's 16B slot in row[] (addrspace(3)
    // offset == low 32 bits of the generic pointer), and matching global
    // byte offset within the row.
    unsigned lds_addr = (unsigned)(size_t)(void*)row + (unsigned)tid * 16u;
    unsigned goff     = (unsigned)tid * 16u;

    // Async DMA the 8KB row into LDS: 2 rounds of 256 lanes x 16B.
    // The instruction offset is added to both the LDS and the global address,
    // so the same vdst/vaddr pair serves both halves via the literal offset.
    asm volatile("global_load_async_to_lds_b128 %0, %1, %2 offset:0"
                 :: "v"(lds_addr), "v"(goff), "s"(in_row) : "memory");
    asm volatile("global_load_async_to_lds_b128 %0, %1, %2 offset:4096"
                 :: "v"(lds_addr), "v"(goff), "s"(in_row) : "memory");
    asm volatile("s_wait_asynccnt 0" ::: "memory");
    __syncthreads();

    // Pull my 8 values out of LDS; kick off gamma loads early so they overlap
    // the reduction. Gamma/out are streamed once -> non-temporal.
    v4f x0 = ((const v4f*)row)[tid];
    v4f x1 = ((const v4f*)row)[tid + NTHREADS];
    v4f g0 = __builtin_nontemporal_load((const v4f*)g_row + tid);
    v4f g1 = __builtin_nontemporal_load((const v4f*)g_row + tid + NTHREADS);

    float s = x0.x * x0.x + x0.y * x0.y + x0.z * x0.z + x0.w * x0.w
            + x1.x * x1.x + x1.y * x1.y + x1.z * x1.z + x1.w * x1.w;

    // wave32 butterfly reduction (warpSize == 32 on gfx1250)
    #pragma unroll
    for (int m = 16; m >= 1; m >>= 1)
        s += __shfl_xor(s, m, 32);

    if ((tid & 31) == 0) partials[tid >> 5] = s;
    __syncthreads();

    float tot = 0.0f;
    #pragma unroll
    for (int i = 0; i < NTHREADS / 32; ++i) tot += partials[i];

    // Uniform per row; redundant per-thread compute is cheaper than another
    // barrier round-trip.
    const float inv = 1.0f / sqrtf(tot * (1.0f / (float)ROW_D) + RMS_EPS);

    v4f o0, o1;
    o0.x = x0.x * inv * g0.x;  o0.y = x0.y * inv * g0.y;
    o0.z = x0.z * inv * g0.z;  o0.w = x0.w * inv * g0.w;
    o1.x = x1.x * inv * g1.x;  o1.y = x1.y * inv * g1.y;
    o1.z = x1.z * inv * g1.z;  o1.w = x1.w * inv * g1.w;

    __builtin_nontemporal_store(o0, (v4f*)o_row + tid);
    __builtin_nontemporal_store(o1, (v4f*)o_row + tid + NTHREADS);
}

extern "C" void kernel_launch(void* const* d_in, const int* in_sizes, int n_in,
                              void* d_out, int out_size, void* d_ws, size_t ws_size,
                              hipStream_t stream)
{
    (void)n_in; (void)d_ws; (void)ws_size;
    const float* in    = (const float*)d_in[0];
    const float* gamma = (const float*)d_in[1];
    float*       out   = (float*)d_out;

    const int rows = out_size / ROW_D;   // 16384
    rmsnorm_f32_kernel<<<rows, NTHREADS, 0, stream>>>(in, gamma, out);
}